// EditEncoder_68058051772850
// MI455X (gfx1250) — compile-verified
//
#include <hip/hip_runtime.h>

typedef float v2f __attribute__((ext_vector_type(2)));
typedef float v8f __attribute__((ext_vector_type(8)));

namespace {
constexpr int kB    = 4096;  // batch
constexpr int kE    = 50;    // max edits
constexpr int kDW   = 512;   // word dim
constexpr int kDH   = 128;   // edit dim / 2
constexpr int kRows = 16;    // batch rows per workgroup
}

__global__ __launch_bounds__(256) void edit_encoder_kernel(
    const float* __restrict__ ins_e,  const float* __restrict__ ins_m,
    const float* __restrict__ insx_e, const float* __restrict__ insx_m,
    const float* __restrict__ del_e,  const float* __restrict__ del_m,
    const float* __restrict__ delx_e, const float* __restrict__ delx_m,
    const float* __restrict__ W, float* __restrict__ out)
{
    // Transposed reduction tile: s_tile[k][row], 512*16*4 = 32 KB.
    // Stage-2 A loads (16 lanes reading rows 0..15 at a fixed k) hit 16
    // consecutive banks -> conflict-free.
    __shared__ float s_tile[kDW * kRows];
    __shared__ float s_m[4][kRows][kE];      // masks, 12.8 KB

    const int tid = threadIdx.x;
    const int b0  = blockIdx.x * kRows;

    // Preload all four mask slices for this 16-row tile.
    for (int i = tid; i < kRows * kE; i += 256) {
        const int r = i / kE, e = i - r * kE;
        const int g = (b0 + r) * kE + e;
        s_m[0][r][e] = ins_m[g];
        s_m[1][r][e] = insx_m[g];
        s_m[2][r][e] = del_m[g];
        s_m[3][r][e] = delx_m[g];
    }
    __syncthreads();

    // Stage-1 mapping: 16 threads per batch row, 32 columns per thread.
    const int row = tid >> 4;                 // 0..15
    const int cl  = tid & 15;                 // 0..15
    const size_t rbase = (size_t)(b0 + row) * kE * kDW;

    // Stage-2 mapping: wave -> 16-col output tile; WMMA 16x16x4 f32 layouts.
    const int wave = tid >> 5;                // 0..7
    const int lane = tid & 31;
    const int nl   = lane & 15;               // N (B/C/D) or M (A) index
    const int half = lane >> 4;               // 0: K={0,1}, 1: K={2,3}
    const int koff = half * 2;
    const int n0   = wave * 16;
    const float* wrow = W + (size_t)(n0 + nl) * kDW;

    const float* pA[2] = {ins_e, del_e};
    const float* pX[2] = {insx_e, delx_e};

    for (int pass = 0; pass < 2; ++pass) {
        const float* base0 = pA[pass] + rbase;
        const float* base1 = pX[pass] + rbase;

        // ---- Stage 1: masked sum over edit axis (bandwidth-bound) ----
        for (int j = 0; j < 8; ++j) {
            const int col = cl * 4 + j * 64;  // 16 threads/row load 256B/step
            const float4* p0 = (const float4*)(base0 + col);
            const float4* p1 = (const float4*)(base1 + col);
            float4 acc = {0.f, 0.f, 0.f, 0.f};
            #pragma unroll 5
            for (int e = 0; e < kE; ++e) {
                const float m0 = s_m[pass * 2 + 0][row][e];
                const float m1 = s_m[pass * 2 + 1][row][e];
                const float4 v0 = p0[e * (kDW / 4)];
                const float4 v1 = p1[e * (kDW / 4)];
                acc.x += v0.x * m0 + v1.x * m1;
                acc.y += v0.y * m0 + v1.y * m1;
                acc.z += v0.z * m0 + v1.z * m1;
                acc.w += v0.w * m0 + v1.w * m1;
            }
            s_tile[(col + 0) * kRows + row] = acc.x;
            s_tile[(col + 1) * kRows + row] = acc.y;
            s_tile[(col + 2) * kRows + row] = acc.z;
            s_tile[(col + 3) * kRows + row] = acc.w;
        }
        __syncthreads();

        // ---- Stage 2: [16,512] @ W^T tile via V_WMMA_F32_16X16X4_F32 ----
        v8f acc = {0.f, 0.f, 0.f, 0.f, 0.f, 0.f, 0.f, 0.f};
        for (int k = 0; k < kDW; k += 4) {
            v2f a;  // A 16x4 f32: lane nl -> M=nl; VGPR0/1 -> K=koff,koff+1
            a.x = s_tile[(k + koff + 0) * kRows + nl];
            a.y = s_tile[(k + koff + 1) * kRows + nl];
            // B 4x16 f32: lane nl -> N=nl; VGPR0/1 -> K=koff,koff+1
            v2f b = *(const v2f*)(wrow + k + koff);
            acc = __builtin_amdgcn_wmma_f32_16x16x4_f32(
                false, a, false, b, (short)0, acc, false, false);
        }

        // C/D layout: VGPR j, lane -> row M = j + half*8, col N = nl.
        #pragma unroll
        for (int j = 0; j < 8; ++j) {
            const int m = j + half * 8;
            out[(size_t)(b0 + m) * (2 * kDH) + pass * kDH + n0 + nl] = acc[j];
        }
        __syncthreads();   // s_tile reused by next pass
    }
}

extern "C" void kernel_launch(void* const* d_in, const int* in_sizes, int n_in,
                              void* d_out, int out_size, void* d_ws, size_t ws_size,
                              hipStream_t stream) {
    (void)in_sizes; (void)n_in; (void)out_size; (void)d_ws; (void)ws_size;
    edit_encoder_kernel<<<kB / kRows, 256, 0, stream>>>(
        (const float*)d_in[0], (const float*)d_in[1],
        (const float*)d_in[2], (const float*)d_in[3],
        (const float*)d_in[4], (const float*)d_in[5],
        (const float*)d_in[6], (const float*)d_in[7],
        (const float*)d_in[8], (float*)d_out);
}